// MemoryFromDecoder_23682449670550
// MI455X (gfx1250) — compile-verified
//
#include <hip/hip_runtime.h>

// Row-wise argmax over R=8192 rows (16x512) of L=32000 fp32 logits.
// softmax is monotone => argmax of logits == reference output.
// Memory-bound: ~1.05 GB streamed once -> ~45us floor at 23.3 TB/s HBM.
//
// MI455X-specific choices:
//  - global_load_b128 th:TH_LOAD_NT (1 GB one-shot stream; don't rinse 192MB L2)
//  - constant trip counts (8000 = 31*256 + 64 peeled tail) so the unroller
//    emits clean clauses of batched NT loads -> high memory-level parallelism
//  - wave32 shuffle reduction (gfx1250 is wave32-only), LDS cross-wave merge

typedef __attribute__((ext_vector_type(4))) float v4f;

constexpr int NT = 256;   // 8 wave32 waves per block

__device__ __forceinline__ void upd(float v, int idx, float& best, int& bestIdx) {
    // Per-thread visit order is strictly increasing index, so strict '>'
    // preserves first-occurrence semantics (matches jnp.argmax).
    if (v > best) { best = v; bestIdx = idx; }
}

template <int L>
__launch_bounds__(NT)
__global__ void argmax_rows_fixed(const float* __restrict__ in,
                                  float* __restrict__ out) {
    const int row = blockIdx.x;
    const int tid = threadIdx.x;

    const v4f* rowv = (const v4f*)(in + (size_t)row * (size_t)L);

    constexpr int nVec = L / 4;          // 8000 float4s per row
    constexpr int FULL = nVec / NT;      // 31 full strided iterations
    constexpr int REM  = nVec % NT;      // 64 leftover float4s

    float best    = -__builtin_huge_valf();
    int   bestIdx = 0;

    // Main loop: constant trip count for every thread -> clean unroll,
    // batched b128 NT loads per clause.
    #pragma unroll 8
    for (int k = 0; k < FULL; ++k) {
        const int i = tid + k * NT;
        v4f v = __builtin_nontemporal_load(rowv + i);   // b128 th:TH_LOAD_NT
        const int base = i * 4;
        upd(v[0], base,     best, bestIdx);
        upd(v[1], base + 1, best, bestIdx);
        upd(v[2], base + 2, best, bestIdx);
        upd(v[3], base + 3, best, bestIdx);
    }
    // Peeled tail: only threads tid < REM take one more float4.
    if (REM > 0 && tid < REM) {
        const int i = tid + FULL * NT;
        v4f v = __builtin_nontemporal_load(rowv + i);
        const int base = i * 4;
        upd(v[0], base,     best, bestIdx);
        upd(v[1], base + 1, best, bestIdx);
        upd(v[2], base + 2, best, bestIdx);
        upd(v[3], base + 3, best, bestIdx);
    }
    // (L % 4 == 0 for the reference shape; template requires it.)
    static_assert(L % 4 == 0, "L must be a multiple of 4");

    // ---- wave32 shuffle reduction; ties -> smaller index wins ----
    #pragma unroll
    for (int off = 16; off > 0; off >>= 1) {
        float ov = __shfl_down(best,    off, 32);
        int   oi = __shfl_down(bestIdx, off, 32);
        if (ov > best || (ov == best && oi < bestIdx)) {
            best = ov; bestIdx = oi;
        }
    }

    // ---- cross-wave reduction via LDS (8 waves) ----
    __shared__ float sVal[NT / 32];
    __shared__ int   sIdx[NT / 32];
    const int lane = tid & 31;
    const int wave = tid >> 5;
    if (lane == 0) { sVal[wave] = best; sIdx[wave] = bestIdx; }
    __syncthreads();

    if (tid == 0) {
        float b  = sVal[0];
        int   bi = sIdx[0];
        #pragma unroll
        for (int w = 1; w < NT / 32; ++w) {
            float ov = sVal[w];
            int   oi = sIdx[w];
            if (ov > b || (ov == b && oi < bi)) { b = ov; bi = oi; }
        }
        out[row] = (float)bi;
    }
}

// Generic-L fallback so kernel_launch stays correct for any shape.
__launch_bounds__(NT)
__global__ void argmax_rows_generic(const float* __restrict__ in,
                                    float* __restrict__ out, int L) {
    const int row = blockIdx.x;
    const int tid = threadIdx.x;

    const float* rowp = in + (size_t)row * (size_t)L;
    const v4f*   rowv = (const v4f*)rowp;
    const int    nVec = L >> 2;

    float best    = -__builtin_huge_valf();
    int   bestIdx = 0;

    #pragma unroll 1
    for (int i = tid; i < nVec; i += NT) {
        v4f v = __builtin_nontemporal_load(rowv + i);
        const int base = i << 2;
        upd(v[0], base,     best, bestIdx);
        upd(v[1], base + 1, best, bestIdx);
        upd(v[2], base + 2, best, bestIdx);
        upd(v[3], base + 3, best, bestIdx);
    }
    #pragma unroll 1
    for (int i = (nVec << 2) + tid; i < L; i += NT) {
        float v = __builtin_nontemporal_load(rowp + i);
        if (v > best) { best = v; bestIdx = i; }
    }

    #pragma unroll
    for (int off = 16; off > 0; off >>= 1) {
        float ov = __shfl_down(best,    off, 32);
        int   oi = __shfl_down(bestIdx, off, 32);
        if (ov > best || (ov == best && oi < bestIdx)) {
            best = ov; bestIdx = oi;
        }
    }

    __shared__ float sVal[NT / 32];
    __shared__ int   sIdx[NT / 32];
    const int lane = tid & 31;
    const int wave = tid >> 5;
    if (lane == 0) { sVal[wave] = best; sIdx[wave] = bestIdx; }
    __syncthreads();

    if (tid == 0) {
        float b  = sVal[0];
        int   bi = sIdx[0];
        #pragma unroll
        for (int w = 1; w < NT / 32; ++w) {
            float ov = sVal[w];
            int   oi = sIdx[w];
            if (ov > b || (ov == b && oi < bi)) { b = ov; bi = oi; }
        }
        out[row] = (float)bi;
    }
}

extern "C" void kernel_launch(void* const* d_in, const int* in_sizes, int n_in,
                              void* d_out, int out_size, void* d_ws, size_t ws_size,
                              hipStream_t stream) {
    const float* in  = (const float*)d_in[0];
    float*       out = (float*)d_out;

    const int R = out_size;                 // 16 * 512 = 8192 rows
    const int L = in_sizes[0] / out_size;   // 32000 logits per row

    if (L == 32000) {
        argmax_rows_fixed<32000><<<dim3(R), dim3(NT), 0, stream>>>(in, out);
    } else {
        argmax_rows_generic<<<dim3(R), dim3(NT), 0, stream>>>(in, out, L);
    }
}